// SurfaceDiceLoss_66760971649521
// MI455X (gfx1250) — compile-verified
//
#include <hip/hip_runtime.h>
#include <hip/hip_bf16.h>
#include <math.h>

#define SMOOTH_ 0.001f
#define B_     2
#define C_     12
#define H_     192
#define W_     192
#define OH_    193
#define OW_    193
#define NPAIR  11
#define TY     8
#define TY1    (TY + 1)
#define NTILE  ((OH_ + TY - 1) / TY)   // 25

// ---- CDNA5 async global->LDS path (guarded; fallback compiles everywhere) ----
#if defined(__has_builtin)
#  if __has_builtin(__builtin_amdgcn_global_load_async_to_lds_b128)
#    define USE_ASYNC_LDS 1
#  endif
#endif

typedef __attribute__((ext_vector_type(4))) int v4i_;

__device__ __forceinline__ void stage_copy16(const float* g, float* l) {
#ifdef USE_ASYNC_LDS
  __builtin_amdgcn_global_load_async_to_lds_b128(
      (__attribute__((address_space(1))) v4i_*)(void*)g,
      (__attribute__((address_space(3))) v4i_*)(void*)l,
      0, 0);
#else
  *(float4*)l = *(const float4*)g;
#endif
}

__device__ __forceinline__ void stage_wait() {
#ifdef USE_ASYNC_LDS
#  if __has_builtin(__builtin_amdgcn_s_wait_asynccnt)
  __builtin_amdgcn_s_wait_asynccnt(0);
#  else
  asm volatile("s_wait_asynccnt 0x0" ::: "memory");
#  endif
#endif
}

// ---- main kernel: one block = (row-tile, pair, batch) ----
__global__ void __launch_bounds__(256)
surface_dice_main(const float* __restrict__ pred,
                  const float* __restrict__ labels,
                  const float* __restrict__ area,
                  float* __restrict__ acc) {
  __shared__ __align__(16) float s_pred[2][TY1][W_];
  __shared__ __align__(16) float s_lab [2][TY1][W_];
  __shared__ float s_area[256];
  __shared__ float s_red[8][4];

  const int t  = threadIdx.x;
  const int y0 = blockIdx.x * TY;
  const int p  = blockIdx.y;   // channel pair 0..10
  const int b  = blockIdx.z;   // batch 0..1

  s_area[t] = area[t];  // blockDim == 256 exactly

  // Stage 2 tensors x 2 channels x TY1 rows x (W/4) 16B quads into LDS.
  const int QPR  = W_ / 4;           // 48 quads per row
  const int NJOB = 2 * 2 * TY1 * QPR;
  for (int j = t; j < NJOB; j += 256) {
    int q   = j % QPR;
    int r   = j / QPR;               // 0..35
    int ts  = r / (2 * TY1);         // 0 = pred, 1 = labels
    int rem = r % (2 * TY1);
    int ch  = rem / TY1;             // 0..1
    int rr  = rem % TY1;             // 0..8 (plane row)
    int gy  = y0 - 1 + rr;           // global input row
    float* lp = ts ? &s_lab[ch][rr][q * 4] : &s_pred[ch][rr][q * 4];
    if (gy >= 0 && gy < H_) {
      const float* base = ts ? labels : pred;
      const float* gp = base +
          ((((size_t)b * C_ + (size_t)(p + ch)) * H_ + (size_t)gy) * W_ + (size_t)(q * 4));
      stage_copy16(gp, lp);
    } else {
      lp[0] = 0.f; lp[1] = 0.f; lp[2] = 0.f; lp[3] = 0.f;  // zero padding rows
    }
  }
  stage_wait();
  __syncthreads();

  float a_num = 0.f, a_den = 0.f, a_bce = 0.f, a_cnt = 0.f;

  for (int idx = t; idx < TY * OW_; idx += 256) {
    int yl = idx / OW_;
    int x  = idx % OW_;              // output col 0..192
    int gy = y0 + yl;                // output row
    if (gy >= OH_) continue;

    float wsum = 0.f, bsum = 0.f;
    int code = 0;
    #pragma unroll
    for (int c = 0; c < 2; ++c) {
      #pragma unroll
      for (int ky = 0; ky < 2; ++ky) {
        int rr  = yl + ky;
        int gyy = gy - 1 + ky;
        bool rin = (gyy >= 0) & (gyy < H_);
        #pragma unroll
        for (int kx = 0; kx < 2; ++kx) {
          int k  = c * 4 + ky * 2 + kx;     // matches reference column order
          int cc = x - 1 + kx;
          bool cin = (cc >= 0) & (cc < W_);
          bool inb = rin & cin;
          float pv = 0.f, lv = 0.f;
          if (inb) { pv = s_pred[c][rr][cc]; lv = s_lab[c][rr][cc]; }
          int bit = (lv > 0.5f) ? 1 : 0;
          code |= bit << k;
          // unfold pads AFTER sigmoid: padded up == 0, not sigmoid(0)
          float up = inb ? (1.f / (1.f + expf(-pv))) : 0.f;
          float d  = up - (float)bit;
          wsum += d * d;
          // bce on zero-padded raw logits/labels (matches reference)
          bsum += fmaxf(pv, 0.f) - pv * lv + log1pf(expf(-fabsf(pv)));
        }
      }
    }
    float wlab = 1.f - sqrtf(wsum) * 0.0625f;
    float la   = s_area[code];
    a_num += 2.f * wlab * la;
    a_den += la;
    if (code == 0 || code == 255) { a_bce += bsum; a_cnt += 1.f; }
  }

  // wave32 shuffle reduction
  #pragma unroll
  for (int off = 16; off > 0; off >>= 1) {
    a_num += __shfl_down(a_num, off, 32);
    a_den += __shfl_down(a_den, off, 32);
    a_bce += __shfl_down(a_bce, off, 32);
    a_cnt += __shfl_down(a_cnt, off, 32);
  }
  int wid = t >> 5, lane = t & 31;
  if (lane == 0) {
    s_red[wid][0] = a_num; s_red[wid][1] = a_den;
    s_red[wid][2] = a_bce; s_red[wid][3] = a_cnt;
  }
  __syncthreads();
  if (t == 0) {
    float n = 0.f, d = 0.f, e = 0.f, c2 = 0.f;
    #pragma unroll
    for (int w = 0; w < 8; ++w) {
      n += s_red[w][0]; d += s_red[w][1]; e += s_red[w][2]; c2 += s_red[w][3];
    }
    float* dst = acc + (size_t)(p * 2 + b) * 4;
    atomicAdd(dst + 0, n);
    atomicAdd(dst + 1, d);
    atomicAdd(dst + 2, e);
    atomicAdd(dst + 3, c2);
  }
}

__global__ void zero_ws(float* ws) {
  int t = threadIdx.x;
  if (t < NPAIR * B_ * 4) ws[t] = 0.f;
}

__global__ void finalize_k(const float* __restrict__ acc,
                           const float* __restrict__ area,
                           float* __restrict__ out) {
  if (threadIdx.x != 0 || blockIdx.x != 0) return;
  const float L  = (float)(OH_ * OW_);   // 37249
  const float a1 = area[1];              // pred_byte == 1 everywhere (proved dead loop)
  float dsum = 0.f, vsum = 0.f;
  for (int b = 0; b < B_; ++b) {
    float num = 0.f, den = 0.f, vol = 0.f;
    for (int p = 0; p < NPAIR; ++p) {
      const float* e = acc + (size_t)(p * 2 + b) * 4;
      num += e[0];
      den += e[1] + L * a1;
      vol += e[2] / (8.f * e[3]);
    }
    dsum += 1.f - (num + SMOOTH_) / (den + SMOOTH_);
    vsum += vol;
  }
  out[0] = dsum * 0.5f + vsum * 0.5f;
}

extern "C" void kernel_launch(void* const* d_in, const int* in_sizes, int n_in,
                              void* d_out, int out_size, void* d_ws, size_t ws_size,
                              hipStream_t stream) {
  const float* pred   = (const float*)d_in[0];
  const float* labels = (const float*)d_in[1];
  const float* area   = (const float*)d_in[2];
  float* ws  = (float*)d_ws;
  float* out = (float*)d_out;

  zero_ws<<<1, 128, 0, stream>>>(ws);
  dim3 grid(NTILE, NPAIR, B_);
  surface_dice_main<<<grid, 256, 0, stream>>>(pred, labels, area, ws);
  finalize_k<<<1, 32, 0, stream>>>(ws, area, out);
}